// GrahSAGE_conv_28836410425907
// MI455X (gfx1250) — compile-verified
//
#include <hip/hip_runtime.h>

typedef __attribute__((ext_vector_type(2))) float v2f;
typedef __attribute__((ext_vector_type(8))) float v8f;
typedef __attribute__((ext_vector_type(4))) int   v4i;

#define THREADS      256
#define EPW          64          // edges per wave in SpMM
#define ROWS_PER_BLK 64          // output rows per GEMM block
#define AS_STRIDE    260         // 256 + 4 pad (bank-conflict avoidance)
#define WT_STRIDE    260

#if defined(__has_builtin)
#if __has_builtin(__builtin_amdgcn_global_load_async_to_lds_b128)
#define HAVE_ASYNC_LDS 1
#endif
#endif

#ifdef HAVE_ASYNC_LDS
typedef __attribute__((address_space(1))) v4i* g_v4i;   // global (AS1) int4*
typedef __attribute__((address_space(3))) v4i* l_v4i;   // LDS (AS3) int4*
#endif

// ---------------------------------------------------------------- zero emb
__global__ __launch_bounds__(THREADS)
void zero_f4(float4* __restrict__ p, int n4) {
    int i = blockIdx.x * blockDim.x + threadIdx.x;
    if (i < n4) p[i] = make_float4(0.f, 0.f, 0.f, 0.f);
}

// ---------------------------------------------------------------- SpMM
// edge_dst is sorted: accumulate runs of equal dst in registers, flush with
// global_atomic_add_f32 only at run boundaries. Lane l owns feature cols
// [4l, 4l+3] (float4 -> 512B coalesced row reads, x is L2-resident).
__global__ __launch_bounds__(THREADS)
void spmm_agg(const float* __restrict__ x,
              const int*   __restrict__ esrc,
              const int*   __restrict__ edst,
              const float* __restrict__ ew,
              float*       __restrict__ emb,
              int E) {
    const int lane = threadIdx.x & 31;
    const int gwave = blockIdx.x * (THREADS >> 5) + (threadIdx.x >> 5);
    int e0 = gwave * EPW;
    if (e0 >= E) return;
    int e1 = e0 + EPW; if (e1 > E) e1 = E;

    float ax = 0.f, ay = 0.f, az = 0.f, aw = 0.f;
    int cur = edst[e0];                       // uniform -> scalar load

    for (int e = e0; e < e1; ++e) {
        int d = edst[e];                      // uniform across wave
        if (d != cur) {
            float* dstp = emb + (size_t)cur * 128 + lane * 4;
            atomicAdd(dstp + 0, ax);
            atomicAdd(dstp + 1, ay);
            atomicAdd(dstp + 2, az);
            atomicAdd(dstp + 3, aw);
            ax = ay = az = aw = 0.f;
            cur = d;
        }
        int   s = esrc[e];                    // uniform
        float w = ew[e];                      // uniform
        const float4 xv = *(const float4*)(x + (size_t)s * 128 + lane * 4);
        ax = fmaf(w, xv.x, ax);
        ay = fmaf(w, xv.y, ay);
        az = fmaf(w, xv.z, az);
        aw = fmaf(w, xv.w, aw);
    }
    float* dstp = emb + (size_t)cur * 128 + lane * 4;
    atomicAdd(dstp + 0, ax);
    atomicAdd(dstp + 1, ay);
    atomicAdd(dstp + 2, az);
    atomicAdd(dstp + 3, aw);
}

// ---------------------------------------------------------------- GEMM
// out[N,128] = relu([x | emb] @ W + bias), W is [256,128] row-major.
// LDS: As = 64x260 h-tile, Wt = 128x260 (W transposed, k-contiguous).
// Wave (of 8) computes one 16-row tile x 64-col half via V_WMMA_F32_16X16X4_F32.
// A-tile staging uses GLOBAL_LOAD_ASYNC_TO_LDS_B128 (memory->LDS DMA);
// out-of-range rows are clamped to N-1 (their products only feed rows that
// are never stored).
extern __shared__ float smem[];

__global__ __launch_bounds__(THREADS)
void gemm_fused(const float* __restrict__ x,
                const float* __restrict__ emb,
                const float* __restrict__ Wg,
                const float* __restrict__ bias,
                float*       __restrict__ out,
                int N) {
    float* As = smem;                               // [64][260]
    float* Wt = smem + ROWS_PER_BLK * AS_STRIDE;    // [128][260] (n-major)
    const int row0 = blockIdx.x * ROWS_PER_BLK;

    // stage h = [x | emb] tile (clamp rows >= N: results masked at store)
    for (int idx = threadIdx.x; idx < ROWS_PER_BLK * 64; idx += THREADS) {
        int r  = idx >> 6;
        int c4 = (idx & 63) << 2;                   // column (0..252, step 4)
        int gr = row0 + r; if (gr >= N) gr = N - 1;
        const float* src = (c4 < 128) ? (x   + (size_t)gr * 128 + c4)
                                      : (emb + (size_t)gr * 128 + (c4 - 128));
        float* dst = As + r * AS_STRIDE + c4;
#ifdef HAVE_ASYNC_LDS
        __builtin_amdgcn_global_load_async_to_lds_b128((g_v4i)src, (l_v4i)dst, 0, 0);
#else
        *(float4*)dst = *(const float4*)src;
#endif
    }
    // stage W transposed (k contiguous per output column)
    for (int idx = threadIdx.x; idx < 256 * 32; idx += THREADS) {
        int k  = idx >> 5;
        int n4 = (idx & 31) << 2;
        float4 w4 = *(const float4*)(Wg + k * 128 + n4);
        Wt[(n4 + 0) * WT_STRIDE + k] = w4.x;
        Wt[(n4 + 1) * WT_STRIDE + k] = w4.y;
        Wt[(n4 + 2) * WT_STRIDE + k] = w4.z;
        Wt[(n4 + 3) * WT_STRIDE + k] = w4.w;
    }
#ifdef HAVE_ASYNC_LDS
    asm volatile("s_wait_asynccnt 0" ::: "memory");
#endif
    __syncthreads();

    const int lane  = threadIdx.x & 31;
    const int wave  = threadIdx.x >> 5;
    const int rbase = (wave >> 1) * 16;             // 4 row tiles
    const int cbase = (wave & 1) * 4;               // 2 col halves (4 tiles each)
    const int m     = lane & 15;
    const int kh    = (lane >> 4) << 1;             // lanes 16-31 hold K+2,K+3

    v8f acc[4] = {};
    const float* Arow = As + (rbase + m) * AS_STRIDE + kh;

    #pragma unroll 4
    for (int k0 = 0; k0 < 256; k0 += 4) {
        v2f a = *(const v2f*)(Arow + k0);           // A frag: 16x4 f32
        #pragma unroll
        for (int ct = 0; ct < 4; ++ct) {
            const float* Bp = Wt + ((cbase + ct) * 16 + m) * WT_STRIDE + k0 + kh;
            v2f b = *(const v2f*)Bp;                // B frag: 4x16 f32
            acc[ct] = __builtin_amdgcn_wmma_f32_16x16x4_f32(
                false, a, false, b, (short)0, acc[ct], false, false);
        }
    }

    // C/D layout: VGPR v -> (M = v + 8*(lane/16), N = lane%16)
    const int rhalf = (lane >> 4) << 3;             // 0 or 8
    const int rowb  = row0 + rbase + rhalf;
    if (row0 + ROWS_PER_BLK <= N) {
        // fast path: whole block in range, no per-element guards
        #pragma unroll
        for (int ct = 0; ct < 4; ++ct) {
            int col = (cbase + ct) * 16 + m;
            float bv = bias[col];
            #pragma unroll
            for (int v = 0; v < 8; ++v) {
                float val = acc[ct][v] + bv;
                out[(size_t)(rowb + v) * 128 + col] = val > 0.f ? val : 0.f;
            }
        }
    } else {
        #pragma unroll
        for (int ct = 0; ct < 4; ++ct) {
            int col = (cbase + ct) * 16 + m;
            float bv = bias[col];
            #pragma unroll
            for (int v = 0; v < 8; ++v) {
                if (rowb + v < N) {
                    float val = acc[ct][v] + bv;
                    out[(size_t)(rowb + v) * 128 + col] = val > 0.f ? val : 0.f;
                }
            }
        }
    }
}

// ---------------------------------------------------------------- launch
extern "C" void kernel_launch(void* const* d_in, const int* in_sizes, int n_in,
                              void* d_out, int out_size, void* d_ws, size_t ws_size,
                              hipStream_t stream) {
    const float* x    = (const float*)d_in[0];
    const int*   esrc = (const int*)  d_in[1];
    const int*   edst = (const int*)  d_in[2];
    const float* ew   = (const float*)d_in[3];
    const float* Wg   = (const float*)d_in[4];
    const float* bias = (const float*)d_in[5];
    float*       out  = (float*)d_out;
    float*       emb  = (float*)d_ws;              // N*128 floats scratch

    const int N = in_sizes[0] / 128;
    const int E = in_sizes[1];

    // 1) zero the aggregation buffer (must happen every replay)
    int n4 = N * 32;                               // N*128/4 float4s
    zero_f4<<<(n4 + THREADS - 1) / THREADS, THREADS, 0, stream>>>((float4*)emb, n4);

    // 2) sorted-COO SpMM with register run-accumulation + f32 atomics
    int waves  = (E + EPW - 1) / EPW;
    int blocks = (waves + (THREADS / 32) - 1) / (THREADS / 32);
    spmm_agg<<<blocks, THREADS, 0, stream>>>(x, esrc, edst, ew, emb, E);

    // 3) WMMA GEMM + bias + ReLU (200 KB dynamic LDS, legal on 320 KB WGP)
    int gblocks = (N + ROWS_PER_BLK - 1) / ROWS_PER_BLK;
    size_t smem_bytes = (size_t)(ROWS_PER_BLK * AS_STRIDE + 128 * WT_STRIDE) * sizeof(float);
    gemm_fused<<<gblocks, THREADS, smem_bytes, stream>>>(x, emb, Wg, bias, out, N);
}